// RHN_36498632082015
// MI455X (gfx1250) — compile-verified
//
#include <hip/hip_runtime.h>
#include <hip/hip_bf16.h>
#include <math.h>
#include <stdint.h>

typedef __attribute__((ext_vector_type(2))) float v2f;
typedef __attribute__((ext_vector_type(8))) float v8f;

#define HID    1024
#define INSZ   512
#define BL     32768   // B*L
#define NSTEP  5
#define KCHUNK 64      // K elements staged per pipeline stage
#define LDSTR  68      // padded LDS row stride (floats) -> conflict-free b64 reads
#define MATSZ  (16 * LDSTR)          // floats per staged 16xKCHUNK matrix
#define BUFSZ  (3 * MATSZ)           // floats per double-buffer half

#define WMMA_F32(A, B, C) \
    __builtin_amdgcn_wmma_f32_16x16x4_f32(false, (A), false, (B), (short)0, (C), false, false)

__device__ __forceinline__ float sigmoid_f(float x) {
    return 1.0f / (1.0f + __expf(-x));
}

// CDNA5 async DMA: GLOBAL_LOAD_ASYNC_TO_LDS_B128 (ASYNCcnt-tracked).
// %0 = per-lane LDS byte offset (VDST), %1 = per-lane 64-bit global address.
__device__ __forceinline__ void async_copy_b128(const float* gsrc, uint32_t lds_off) {
    asm volatile("global_load_async_to_lds_b128 %0, %1, off"
                 :
                 : "v"(lds_off), "v"(gsrc)
                 : "memory");
}
__device__ __forceinline__ void wait_async0() {
    asm volatile("s_wait_asynccnt 0x0" ::: "memory");
}

// ---------------------------------------------------------------------------
// Kernel 1: fused GRU input step (h0 = 0).
// Block: 256 threads (8 waves), tile 256(M) x 16(N); each wave owns two
// 16x16 M-tiles. The three gate-weight strips (r,z,n rows of w_ih) stream
// through double-buffered LDS via async global->LDS DMA.
// ---------------------------------------------------------------------------
__global__ __launch_bounds__(256) void rhn_gru_kernel(
    const float* __restrict__ x,      // [BL, INSZ]
    const float* __restrict__ w_ih,   // [3*HID, INSZ]
    const float* __restrict__ b_ih,   // [3*HID]
    const float* __restrict__ b_hh,   // [3*HID]
    float* __restrict__ h)            // [BL, HID]
{
    __shared__ float ldsB[2][3][MATSZ];

    const int tid  = threadIdx.x;
    const int lane = tid & 31;
    const int wave = tid >> 5;
    const int l16  = lane & 15;
    const int hl   = lane >> 4;
    const int koff = hl * 2;

    const int n0 = blockIdx.x * 16;
    const int m0 = blockIdx.y * 256 + wave * 32;

    // Cooperative weight DMA: thread -> (row = tid>>4, float4 column = tid&15)
    const int lrow = tid >> 4;
    const int lkv  = (tid & 15) * 4;
    const float* gR = w_ih + (size_t)(n0 + lrow) * INSZ + lkv;
    const float* gZ = w_ih + (size_t)(HID   + n0 + lrow) * INSZ + lkv;
    const float* gN = w_ih + (size_t)(2*HID + n0 + lrow) * INSZ + lkv;

    const uint32_t ldsBase =
        (uint32_t)(uintptr_t)&ldsB[0][0][0] + (uint32_t)((lrow * LDSTR + lkv) * 4);

    const float* a0 = x + (size_t)(m0 + l16) * INSZ + koff;        // M-tile 0
    const float* a1 = x + (size_t)(m0 + 16 + l16) * INSZ + koff;   // M-tile 1

    v8f aR0 = {}, aR1 = {}, aZ0 = {}, aZ1 = {}, aN0 = {}, aN1 = {};

    // Pipeline prologue: DMA chunk 0 into buffer 0.
    async_copy_b128(gR, ldsBase);
    async_copy_b128(gZ, ldsBase + MATSZ * 4);
    async_copy_b128(gN, ldsBase + 2 * MATSZ * 4);
    wait_async0();
    __syncthreads();

    const int NCH = INSZ / KCHUNK;   // 8
    for (int c = 0; c < NCH; ++c) {
        const int cur = c & 1;
        if (c + 1 < NCH) {  // DMA next chunk while this chunk computes
            const uint32_t nb = ldsBase + (uint32_t)((cur ^ 1) * BUFSZ * 4);
            async_copy_b128(gR + (c + 1) * KCHUNK, nb);
            async_copy_b128(gZ + (c + 1) * KCHUNK, nb + MATSZ * 4);
            async_copy_b128(gN + (c + 1) * KCHUNK, nb + 2 * MATSZ * 4);
        }
        const float* sR = &ldsB[cur][0][l16 * LDSTR + koff];
        const float* sZ = &ldsB[cur][1][l16 * LDSTR + koff];
        const float* sN = &ldsB[cur][2][l16 * LDSTR + koff];
        const float* ca0 = a0 + c * KCHUNK;
        const float* ca1 = a1 + c * KCHUNK;
#pragma unroll
        for (int s = 0; s < KCHUNK / 4; ++s) {
            v2f av0 = *(const v2f*)(ca0 + s * 4);
            v2f av1 = *(const v2f*)(ca1 + s * 4);
            v2f fr  = *(const v2f*)(sR + s * 4);
            v2f fz  = *(const v2f*)(sZ + s * 4);
            v2f fn  = *(const v2f*)(sN + s * 4);
            aR0 = WMMA_F32(av0, fr, aR0);  aR1 = WMMA_F32(av1, fr, aR1);
            aZ0 = WMMA_F32(av0, fz, aZ0);  aZ1 = WMMA_F32(av1, fz, aZ1);
            aN0 = WMMA_F32(av0, fn, aN0);  aN1 = WMMA_F32(av1, fn, aN1);
        }
        wait_async0();       // own DMA done before signaling
        __syncthreads();     // all waves' DMA visible; fences WAR on buffers
    }

    const int col = n0 + l16;
    const float b_r  = b_ih[col]       + b_hh[col];
    const float b_z  = b_ih[HID + col] + b_hh[HID + col];
    const float b_in = b_ih[2*HID + col];
    const float b_hn = b_hh[2*HID + col];

#pragma unroll
    for (int v = 0; v < 8; ++v) {
        const int m = m0 + v + hl * 8;
        {
            float r  = sigmoid_f(aR0[v] + b_r);
            float z  = sigmoid_f(aZ0[v] + b_z);
            float nn = tanhf(aN0[v] + b_in + r * b_hn);
            h[(size_t)m * HID + col] = (1.0f - z) * nn;
        }
        {
            float r  = sigmoid_f(aR1[v] + b_r);
            float z  = sigmoid_f(aZ1[v] + b_z);
            float nn = tanhf(aN1[v] + b_in + r * b_hn);
            h[(size_t)(m + 16) * HID + col] = (1.0f - z) * nn;
        }
    }
}

// ---------------------------------------------------------------------------
// Kernel 2: one highway micro-step, three fused GEMMs + gating.
//   out = tanh(hWH^T+bH) * sig(hWT^T+bT) + h * (1 - sig(hWC^T+bC))
// Same async-DMA double-buffered weight pipeline, 2 M-tiles per wave.
// ---------------------------------------------------------------------------
__global__ __launch_bounds__(256) void rhn_micro_kernel(
    const float* __restrict__ h,    // [BL, HID]
    const float* __restrict__ WH, const float* __restrict__ bH,
    const float* __restrict__ WT, const float* __restrict__ bT,
    const float* __restrict__ WC, const float* __restrict__ bC,
    float* __restrict__ out)        // [BL, HID]
{
    __shared__ float ldsB[2][3][MATSZ];

    const int tid  = threadIdx.x;
    const int lane = tid & 31;
    const int wave = tid >> 5;
    const int l16  = lane & 15;
    const int hl   = lane >> 4;
    const int koff = hl * 2;

    const int n0 = blockIdx.x * 16;
    const int m0 = blockIdx.y * 256 + wave * 32;

    const int lrow = tid >> 4;
    const int lkv  = (tid & 15) * 4;
    const float* gH = WH + (size_t)(n0 + lrow) * HID + lkv;
    const float* gT = WT + (size_t)(n0 + lrow) * HID + lkv;
    const float* gC = WC + (size_t)(n0 + lrow) * HID + lkv;

    const uint32_t ldsBase =
        (uint32_t)(uintptr_t)&ldsB[0][0][0] + (uint32_t)((lrow * LDSTR + lkv) * 4);

    const float* a0 = h + (size_t)(m0 + l16) * HID + koff;
    const float* a1 = h + (size_t)(m0 + 16 + l16) * HID + koff;

    v8f aH0 = {}, aH1 = {}, aT0 = {}, aT1 = {}, aC0 = {}, aC1 = {};

    async_copy_b128(gH, ldsBase);
    async_copy_b128(gT, ldsBase + MATSZ * 4);
    async_copy_b128(gC, ldsBase + 2 * MATSZ * 4);
    wait_async0();
    __syncthreads();

    const int NCH = HID / KCHUNK;   // 16
    for (int c = 0; c < NCH; ++c) {
        const int cur = c & 1;
        if (c + 1 < NCH) {
            const uint32_t nb = ldsBase + (uint32_t)((cur ^ 1) * BUFSZ * 4);
            async_copy_b128(gH + (c + 1) * KCHUNK, nb);
            async_copy_b128(gT + (c + 1) * KCHUNK, nb + MATSZ * 4);
            async_copy_b128(gC + (c + 1) * KCHUNK, nb + 2 * MATSZ * 4);
        }
        const float* sH = &ldsB[cur][0][l16 * LDSTR + koff];
        const float* sT = &ldsB[cur][1][l16 * LDSTR + koff];
        const float* sC = &ldsB[cur][2][l16 * LDSTR + koff];
        const float* ca0 = a0 + c * KCHUNK;
        const float* ca1 = a1 + c * KCHUNK;
#pragma unroll
        for (int s = 0; s < KCHUNK / 4; ++s) {
            v2f av0 = *(const v2f*)(ca0 + s * 4);
            v2f av1 = *(const v2f*)(ca1 + s * 4);
            v2f fh  = *(const v2f*)(sH + s * 4);
            v2f ft  = *(const v2f*)(sT + s * 4);
            v2f fc  = *(const v2f*)(sC + s * 4);
            aH0 = WMMA_F32(av0, fh, aH0);  aH1 = WMMA_F32(av1, fh, aH1);
            aT0 = WMMA_F32(av0, ft, aT0);  aT1 = WMMA_F32(av1, ft, aT1);
            aC0 = WMMA_F32(av0, fc, aC0);  aC1 = WMMA_F32(av1, fc, aC1);
        }
        wait_async0();
        __syncthreads();
    }

    const int col = n0 + l16;
    const float bh = bH[col];
    const float bt = bT[col];
    const float bc = bC[col];

#pragma unroll
    for (int v = 0; v < 8; ++v) {
        const int m = m0 + v + hl * 8;
        {
            const size_t idx = (size_t)m * HID + col;
            float hv  = h[idx];
            float hhv = tanhf(aH0[v] + bh);
            float tv  = sigmoid_f(aT0[v] + bt);
            float cv  = sigmoid_f(aC0[v] + bc);
            out[idx] = hhv * tv + hv * (1.0f - cv);
        }
        {
            const size_t idx = (size_t)(m + 16) * HID + col;
            float hv  = h[idx];
            float hhv = tanhf(aH1[v] + bh);
            float tv  = sigmoid_f(aT1[v] + bt);
            float cv  = sigmoid_f(aC1[v] + bc);
            out[idx] = hhv * tv + hv * (1.0f - cv);
        }
    }
}

// ---------------------------------------------------------------------------
// Launch: GRU -> ws, then 5 micro-steps ping-ponging ws <-> d_out so the
// final (odd-count) step lands in d_out. Graph-capture safe.
// ---------------------------------------------------------------------------
extern "C" void kernel_launch(void* const* d_in, const int* in_sizes, int n_in,
                              void* d_out, int out_size, void* d_ws, size_t ws_size,
                              hipStream_t stream) {
    (void)in_sizes; (void)n_in; (void)out_size; (void)ws_size;

    const float* x    = (const float*)d_in[0];
    const float* w_ih = (const float*)d_in[1];
    // d_in[2] = w_hh: zero contribution (h0 == 0).
    const float* b_ih = (const float*)d_in[3];
    const float* b_hh = (const float*)d_in[4];
    const float* WH   = (const float*)d_in[5];
    const float* bH   = (const float*)d_in[6];
    const float* WT   = (const float*)d_in[7];
    const float* bT   = (const float*)d_in[8];
    const float* WC   = (const float*)d_in[9];
    const float* bC   = (const float*)d_in[10];

    float* bufA = (float*)d_ws;     // 128 MB scratch
    float* bufO = (float*)d_out;

    dim3 grid(HID / 16, BL / 256);  // (64, 128)
    dim3 block(256);                // 8 wave32 waves

    rhn_gru_kernel<<<grid, block, 0, stream>>>(x, w_ih, b_ih, b_hh, bufA);

    const float* cur = bufA;
    for (int t = 0; t < NSTEP; ++t) {
        float* dst = (t & 1) ? bufA : bufO;   // t = 0,2,4 -> d_out (final at t=4)
        rhn_micro_kernel<<<grid, block, 0, stream>>>(
            cur,
            WH + (size_t)t * HID * HID, bH + (size_t)t * HID,
            WT + (size_t)t * HID * HID, bT + (size_t)t * HID,
            WC + (size_t)t * HID * HID, bC + (size_t)t * HID,
            dst);
        cur = dst;
    }
}